// Loss_5506148073791
// MI455X (gfx1250) — compile-verified
//
#include <hip/hip_runtime.h>
#include <hip/hip_bf16.h>

#define NROW 8192
#define BITL 32
#define NCLS 100
#define ALPHA 0.1f
#define BEITA 1.0f
#define GAMMA 0.1f

typedef __attribute__((ext_vector_type(2))) float v2f;
typedef __attribute__((ext_vector_type(8))) float v8f;

// ---- workspace layout (bytes) ----
// [0, 32768)        : int labels[8192]
// [32768, 33280)    : int counts[128]  (only 100 used)
// [33280, 33288)    : float scales[2]  {scale_pos = s/s1, scale_neg = s/s0}
// [33288, 33320)    : double acc[4]    {pair_sum, ce_sum, msq_sum, bal_sum}

__device__ __forceinline__ float waveSum(float v) {
#pragma unroll
  for (int m = 16; m >= 1; m >>= 1) v += __shfl_xor(v, m, 32);
  return v;
}
__device__ __forceinline__ float waveMax(float v) {
#pragma unroll
  for (int m = 16; m >= 1; m >>= 1) v = fmaxf(v, __shfl_xor(v, m, 32));
  return v;
}

__global__ void dpsh_init_kernel(int* counts, float* scales, double* acc) {
  int t = threadIdx.x;
  if (t < 128) counts[t] = 0;
  if (t < 2) scales[t] = 0.0f;
  if (t < 4) acc[t] = 0.0;
}

// labels[i] = argwhere(y[i]==1); histogram class counts
__global__ void dpsh_label_kernel(const float* __restrict__ y, int* __restrict__ labels,
                                  int* __restrict__ counts) {
  int row = blockIdx.x * blockDim.x + threadIdx.x;
  if (row >= NROW) return;
  int lab = 0;
  const float* yr = y + (size_t)row * NCLS;
#pragma unroll 4
  for (int j = 0; j < NCLS; ++j)
    if (yr[j] > 0.5f) lab = j;
  labels[row] = lab;
  atomicAdd(&counts[lab], 1);
}

// s1 = sum(count^2) - n ; s0 = n^2 - sum(count^2) ; guards ; s = s0+s1
__global__ void dpsh_scale_kernel(const int* __restrict__ counts, float* __restrict__ scales) {
  __shared__ long long part[128];
  int t = threadIdx.x;
  long long c = (t < NCLS) ? (long long)counts[t] : 0;
  part[t] = c * c;
  __syncthreads();
  for (int ofs = 64; ofs >= 1; ofs >>= 1) {
    if (t < ofs) part[t] += part[t + ofs];
    __syncthreads();
  }
  if (t == 0) {
    double n = (double)NROW;
    double sumsq = (double)part[0];
    double s1 = sumsq - n;
    double s0 = n * n - sumsq;
    if (s0 == 0.0) s0 = 1.0;
    if (s1 == 0.0) s1 = 1.0;
    double s = s0 + s1;
    scales[0] = (float)(s / s1);  // sim > 0
    scales[1] = (float)(s / s0);  // sim == 0
  }
}

// Pairwise likelihood loss: tile-wise u @ u^T via V_WMMA_F32_16X16X4_F32.
// Each wave: one 16-row panel vs 4 consecutive 16-col tiles (64 cols).
// grid: 8192 blocks x 256 threads (8 waves) -> 65536 waves = 512 row-tiles x 128 col-groups.
__global__ void __launch_bounds__(256) dpsh_pair_kernel(
    const float* __restrict__ u, const int* __restrict__ labels,
    const float* __restrict__ scales, double* __restrict__ acc) {
  const int lane = threadIdx.x & 31;
  const int warp = threadIdx.x >> 5;
  const int waveId = blockIdx.x * 8 + warp;
  const int rt = waveId >> 7;          // 0..511 row tile
  const int cg = waveId & 127;         // 0..127 col group (4 tiles)
  const int i0 = rt * 16;
  const int j0base = cg * 64;

  const int half = lane >> 4;          // K sub-block select
  const int l = lane & 15;             // row/col within tile

  // Preload A panel: A[m=l][k], VGPR0=K(2*half), VGPR1=K(2*half+1), per k-step of 4
  v2f a[8];
  const int rowA = i0 + l;
#pragma unroll
  for (int s = 0; s < 8; ++s) {
    int kb = 4 * s + 2 * half;
    a[s] = *(const v2f*)(u + (size_t)rowA * BITL + kb);
  }

  // labels for the 8 output rows this lane owns (C layout: m = v + 8*half)
  int labI[8];
#pragma unroll
  for (int v = 0; v < 8; ++v) labI[v] = labels[i0 + v + 8 * half];

  const float scPos = scales[0];
  const float scNeg = scales[1];

  float accf = 0.0f;

#pragma unroll 1
  for (int jt = 0; jt < 4; ++jt) {
    const int j0 = j0base + jt * 16;
    const int rowB = j0 + l;           // B[k][n] = u[j0+n][k]
    v8f c = {0.f, 0.f, 0.f, 0.f, 0.f, 0.f, 0.f, 0.f};
#pragma unroll
    for (int s = 0; s < 8; ++s) {
      int kb = 4 * s + 2 * half;
      v2f b = *(const v2f*)(u + (size_t)rowB * BITL + kb);
      c = __builtin_amdgcn_wmma_f32_16x16x4_f32(false, a[s], false, b,
                                                (short)0, c, false, false);
    }
    const int labJ = labels[j0 + l];   // column n = lane&15
#pragma unroll
    for (int v = 0; v < 8; ++v) {
      int i = i0 + v + 8 * half;
      int j = j0 + l;
      if (i == j) continue;            // diag removed in forward value
      float theta = 0.5f * c[v];
      bool sim = (labI[v] == labJ);
      float sp = fmaxf(theta, 0.0f) + log1pf(expf(-fabsf(theta)));
      float pair = sim ? (sp - theta) * scPos : sp * scNeg;
      accf += pair;
    }
  }

  // block reduce -> one f64 atomic per block
  __shared__ double wsum[8];
  float ws = waveSum(accf);
  if (lane == 0) wsum[warp] = (double)ws;
  __syncthreads();
  if (threadIdx.x == 0) {
    double s = 0.0;
#pragma unroll
    for (int w = 0; w < 8; ++w) s += wsum[w];
    atomicAdd(&acc[0], s);
  }
}

// Cross-entropy: loss_i = logsumexp(Y[i]) - Y[i][label_i]; one wave per row.
__global__ void __launch_bounds__(256) dpsh_ce_kernel(
    const float* __restrict__ Y, const int* __restrict__ labels,
    double* __restrict__ acc) {
  const int lane = threadIdx.x & 31;
  const int warp = threadIdx.x >> 5;
  const int row = blockIdx.x * 8 + warp;
  const float* yr = Y + (size_t)row * NCLS;

  float mx = -3.402823466e+38f;
  for (int j = lane; j < NCLS; j += 32) mx = fmaxf(mx, yr[j]);
  mx = waveMax(mx);
  float se = 0.0f;
  for (int j = lane; j < NCLS; j += 32) se += expf(yr[j] - mx);
  se = waveSum(se);
  float loss = 0.0f;
  if (lane == 0) loss = (mx + logf(se)) - yr[labels[row]];

  __shared__ double wsum[8];
  if (lane == 0) wsum[warp] = (double)loss;
  __syncthreads();
  if (threadIdx.x == 0) {
    double s = 0.0;
#pragma unroll
    for (int w = 0; w < 8; ++w) s += wsum[w];
    atomicAdd(&acc[1], s);
  }
}

// Quantization (row-mean squared) + balancing ((u - sign(u))^2); one wave per row.
__global__ void __launch_bounds__(256) dpsh_qb_kernel(
    const float* __restrict__ u, double* __restrict__ acc) {
  const int lane = threadIdx.x & 31;
  const int warp = threadIdx.x >> 5;
  const int row = blockIdx.x * 8 + warp;
  float val = u[(size_t)row * BITL + lane];   // BIT==32 == wave width

  float msum = waveSum(val);
  float sgn = (val > 0.0f) ? 1.0f : ((val < 0.0f) ? -1.0f : 0.0f);
  float d = val - sgn;
  float bsum = waveSum(d * d);

  __shared__ double wm[8], wb[8];
  if (lane == 0) {
    float m = msum * (1.0f / BITL);
    wm[warp] = (double)(m * m);
    wb[warp] = (double)bsum;
  }
  __syncthreads();
  if (threadIdx.x == 0) {
    double sm = 0.0, sb = 0.0;
#pragma unroll
    for (int w = 0; w < 8; ++w) { sm += wm[w]; sb += wb[w]; }
    atomicAdd(&acc[2], sm);
    atomicAdd(&acc[3], sb);
  }
}

__global__ void dpsh_final_kernel(const double* __restrict__ acc, float* __restrict__ out) {
  double n = (double)NROW;
  double count = n * (n - 1.0) * 0.5;
  double likelihood = acc[0] / 2.0 / count;
  double ce = acc[1] / n;
  double quant = (double)ALPHA * (acc[2] / n);
  double bal = (double)GAMMA * (acc[3] / (n * (double)BITL));
  out[0] = (float)(likelihood + (double)BEITA * ce + quant + bal);
}

extern "C" void kernel_launch(void* const* d_in, const int* in_sizes, int n_in,
                              void* d_out, int out_size, void* d_ws, size_t ws_size,
                              hipStream_t stream) {
  const float* u = (const float*)d_in[0];   // [8192, 32]
  const float* Y = (const float*)d_in[1];   // [8192, 100]
  const float* y = (const float*)d_in[2];   // [8192, 100] one-hot
  float* out = (float*)d_out;

  char* ws = (char*)d_ws;
  int*    labels = (int*)(ws);
  int*    counts = (int*)(ws + 32768);
  float*  scales = (float*)(ws + 33280);
  double* acc    = (double*)(ws + 33288);

  dpsh_init_kernel<<<1, 128, 0, stream>>>(counts, scales, acc);
  dpsh_label_kernel<<<NROW / 256, 256, 0, stream>>>(y, labels, counts);
  dpsh_scale_kernel<<<1, 128, 0, stream>>>(counts, scales);
  // 512 row-tiles * 128 col-groups = 65536 waves / 8 waves per block
  dpsh_pair_kernel<<<8192, 256, 0, stream>>>(u, labels, scales, acc);
  dpsh_ce_kernel<<<NROW / 8, 256, 0, stream>>>(Y, labels, acc);
  dpsh_qb_kernel<<<NROW / 8, 256, 0, stream>>>(u, acc);
  dpsh_final_kernel<<<1, 1, 0, stream>>>(acc, out);
}